// SageEmbedding_88811333746889
// MI455X (gfx1250) — compile-verified
//
#include <hip/hip_runtime.h>
#include <hip/hip_bf16.h>
#include <stdint.h>

// ---------------------------------------------------------------------------
// DenseSAGE x2, fused, fp32 via V_WMMA_F32_16X16X4_F32 (gfx1250 / CDNA5)
// One wave32 handles one batch (16 nodes == one WMMA M-tile).
// Weights pre-packed into B-fragment order (one b64/lane, coalesced).
// x tile staged through the Tensor Data Mover when the builtin is available.
// ---------------------------------------------------------------------------

typedef __attribute__((ext_vector_type(2))) float    v2f;
typedef __attribute__((ext_vector_type(4))) float    f4;
typedef __attribute__((ext_vector_type(8))) float    v8f;
typedef __attribute__((ext_vector_type(4))) unsigned u32x4;
typedef __attribute__((ext_vector_type(4))) int      i32x4;
typedef __attribute__((ext_vector_type(8))) int      i32x8;

#define NND   16     // nodes per batch
#define FIN   64
#define FHID  128
#define FOUT  128
#define WAVES 2      // waves (batches) per block

#define XS_STRIDE 72    // 2*stride mod 64 == 16 -> half-waves hit disjoint banks
#define HS_STRIDE 136
#define XS_SZ (16 * XS_STRIDE)           // 1152 floats
#define HS_SZ (16 * HS_STRIDE)           // 2176 floats
#define REGION1 (2 * XS_SZ)              // xs + xin; reused as hin (2176 <= 2304)
#define PER_WAVE (REGION1 + HS_SZ + 32)  // + keep[16] + deg[16]

#define PW1_PAIRS (8 * 32 * 32)          // ntile x kstep x lane  (layer 1, K=128)
#define PW2_PAIRS (8 * 64 * 32)          // ntile x kstep x lane  (layer 2, K=256)

#if __has_builtin(__builtin_amdgcn_tensor_load_to_lds) && \
    __has_builtin(__builtin_amdgcn_s_wait_tensorcnt)
#define HAVE_TDM 1
#endif

__global__ void flags_zero_kernel(unsigned* __restrict__ f, int n) {
  int i = blockIdx.x * blockDim.x + threadIdx.x;
  if (i < n) f[i] = 0u;
}

__global__ void flags_scatter_kernel(const int* __restrict__ e,
                                     unsigned* __restrict__ f, int ne, int n) {
  int i = blockIdx.x * blockDim.x + threadIdx.x;
  if (i < ne) {
    int v = e[i];
    if (v >= 0 && v < n) f[v] = 1u;   // idempotent store, no atomic needed
  }
}

// Pack [Wr;Wo] into per-lane WMMA B-fragment pairs:
//   PW[((nt*KSTEPS + ks)*32 + lane)] = { B[4ks+2h][nt*16+c], B[4ks+2h+1][nt*16+c] }
__global__ void pack_weights_kernel(const float* __restrict__ Wr1,
                                    const float* __restrict__ Wo1,
                                    const float* __restrict__ Wr2,
                                    const float* __restrict__ Wo2,
                                    v2f* __restrict__ PW1, v2f* __restrict__ PW2) {
  int tid = blockIdx.x * blockDim.x + threadIdx.x;
  if (tid < PW1_PAIRS) {
    int lane = tid & 31, ks = (tid >> 5) & 31, nt = tid >> 10;
    int hlf = lane >> 4;
    int c = nt * 16 + (lane & 15);
    int k0 = 4 * ks + 2 * hlf;                       // even; pairs never straddle 64
    const float* w = (k0 < 64) ? (Wr1 + k0 * FHID) : (Wo1 + (k0 - 64) * FHID);
    v2f v; v.x = w[c]; v.y = w[FHID + c];
    PW1[tid] = v;
  } else if (tid < PW1_PAIRS + PW2_PAIRS) {
    int t = tid - PW1_PAIRS;
    int lane = t & 31, ks = (t >> 5) & 63, nt = t >> 11;
    int hlf = lane >> 4;
    int c = nt * 16 + (lane & 15);
    int k0 = 4 * ks + 2 * hlf;                       // even; pairs never straddle 128
    const float* w = (k0 < 128) ? (Wr2 + k0 * FOUT) : (Wo2 + (k0 - 128) * FOUT);
    v2f v; v.x = w[c]; v.y = w[FOUT + c];
    PW2[t] = v;
  }
}

__global__ __launch_bounds__(WAVES * 32)
void sage_fused_kernel(const float* __restrict__ x,
                       const float* __restrict__ adjp,
                       const float* __restrict__ b1, const float* __restrict__ b2,
                       const v2f* __restrict__ PW1, const v2f* __restrict__ PW2,
                       const unsigned* __restrict__ flags,
                       float* __restrict__ out, int nb) {
  __shared__ float smem[WAVES * PER_WAVE];

  const int wave = threadIdx.x >> 5;
  const int lane = threadIdx.x & 31;
  const int half = lane >> 4;    // K/M-half select per WMMA layouts
  const int col  = lane & 15;    // B/C/D column for this lane
  const int row  = col;          // A row for this lane
  int b = blockIdx.x * WAVES + wave;
  if (b >= nb) b = nb - 1;       // duplicate last batch (identical writes, benign)

  float* base  = smem + wave * PER_WAVE;
  float* xs    = base;                     // x   tile  [16 x 64], stride 72
  float* xin   = base + XS_SZ;             // agg1/deg  [16 x 64], stride 72
  float* hin   = base;                     // agg2/deg  [16 x 128], stride 136 (reuse)
  float* hs    = base + REGION1;           // h   tile  [16 x 128], stride 136
  float* keepS = base + REGION1 + HS_SZ;   // keep[16]
  float* degS  = keepS + 16;               // deg[16]

  // ---- kick off x tile staging [16 x 64] -> LDS (stride 72 via TDM padding) ----
#ifdef HAVE_TDM
  {
    unsigned lds_off = (unsigned)(uintptr_t)xs;     // flat LDS ptr low bits = LDS byte offset
    unsigned long long ga =
        (unsigned long long)(uintptr_t)(x + (size_t)b * NND * FIN);
    u32x4 g0;
    g0[0] = 1u;                                     // count=1, user descriptor
    g0[1] = lds_off;                                // lds_addr
    g0[2] = (unsigned)(ga & 0xFFFFFFFFu);           // global_addr[31:0]
    g0[3] = (unsigned)((ga >> 32) & 0x01FFFFFFu)    // global_addr[56:32]
            | 0x80000000u;                          // type = 2 ("image")
    i32x8 g1;
    g1[0] = (int)((2u << 16)                        // data_size = 4B
                  | (1u << 20)                      // pad_enable
                  | (5u << 22)                      // pad_interval: every 64 DWORDs
                  | (7u << 25));                    // pad_amount: 8 DWORDs
    g1[1] = (int)(64u << 16);                       // tensor_dim0 = 64
    g1[2] = (int)(16u << 16);                       // tensor_dim1 = 16
    g1[3] = (int)(64u << 16);                       // tile_dim0 = 64
    g1[4] = 16;                                     // tile_dim1 = 16
    g1[5] = 64;                                     // tensor_dim0_stride = 64
    g1[6] = 0; g1[7] = 0;
    i32x4 g2 = {0, 0, 0, 0}, g3 = {0, 0, 0, 0};
#if __clang_major__ >= 23
    i32x8 g4 = {0, 0, 0, 0, 0, 0, 0, 0};
    __builtin_amdgcn_tensor_load_to_lds(g0, g1, g2, g3, g4, 0);
#else
    __builtin_amdgcn_tensor_load_to_lds(g0, g1, g2, g3, 0);
#endif
  }
#else
  {
    const f4* xg = (const f4*)(x + (size_t)b * NND * FIN);
#pragma unroll
    for (int t = 0; t < 8; ++t) {
      int idx = lane + 32 * t;        // 256 float4 total
      int r = idx >> 4, c4 = idx & 15;
      f4 val = xg[r * 16 + c4];
      *(f4*)(xs + r * XS_STRIDE + c4 * 4) = val;
    }
  }
#endif

  // ---- node keep flags -> LDS ----
  if (half == 0) keepS[col] = flags[b * NND + col] ? 1.0f : 0.0f;
  __syncthreads();

  // ---- degree (clamped to 1) ----
  if (half == 0) {
    float s = 0.0f;
#pragma unroll
    for (int j = 0; j < 16; ++j) s += keepS[j] * adjp[col * 16 + j];
    degS[col] = fmaxf(s * keepS[col], 1.0f);
  }

  // ---- adjacency A-fragments: a[k][v] = adj[row][4k + 2*half + v] ----
  float adjA[4][2];
  {
    float kr = keepS[row];
#pragma unroll
    for (int k = 0; k < 4; ++k)
#pragma unroll
      for (int v = 0; v < 2; ++v) {
        int j = 4 * k + 2 * half + v;
        adjA[k][v] = kr * keepS[j] * adjp[row * 16 + j];
      }
  }
#ifdef HAVE_TDM
  __builtin_amdgcn_s_wait_tensorcnt(0);   // x tile resident in LDS
#endif
  __syncthreads();

  float rdeg[8];
#pragma unroll
  for (int v = 0; v < 8; ++v)
    rdeg[v] = __builtin_amdgcn_rcpf(degS[v + 8 * half]);   // deg >= 1: rcp safe

  // ================= layer 1: agg1 = adj @ x, xin = agg1/deg ==============
#pragma unroll
  for (int nt = 0; nt < 4; ++nt) {
    v8f acc = {};
#pragma unroll
    for (int k = 0; k < 4; ++k) {
      int k0 = 4 * k + 2 * half;
      v2f a  = {adjA[k][0], adjA[k][1]};
      v2f bf = {xs[(k0 + 0) * XS_STRIDE + nt * 16 + col],
                xs[(k0 + 1) * XS_STRIDE + nt * 16 + col]};
      acc = __builtin_amdgcn_wmma_f32_16x16x4_f32(false, a, false, bf,
                                                  (short)0, acc, false, false);
    }
#pragma unroll
    for (int v = 0; v < 8; ++v)
      xin[(v + 8 * half) * XS_STRIDE + nt * 16 + col] = acc[v] * rdeg[v];
  }
  __syncthreads();

  // ============ layer 1 GEMM: h = relu([xin|x] @ [Wr1;Wo1] + b1) ==========
  for (int nt = 0; nt < 8; ++nt) {
    float bias = b1[nt * 16 + col];
    v8f acc;
#pragma unroll
    for (int v = 0; v < 8; ++v) acc[v] = bias;
    const v2f* p = PW1 + (nt * 32) * 32 + lane;      // packed B-fragments
    for (int ks = 0; ks < 16; ++ks) {                // K = 64 (neighbor path)
      v2f a  = *(const v2f*)(xin + row * XS_STRIDE + 4 * ks + 2 * half);
      v2f bf = p[ks * 32];
      acc = __builtin_amdgcn_wmma_f32_16x16x4_f32(false, a, false, bf,
                                                  (short)0, acc, false, false);
    }
    for (int ks = 16; ks < 32; ++ks) {               // K = 64 (root path)
      v2f a  = *(const v2f*)(xs + row * XS_STRIDE + 4 * (ks - 16) + 2 * half);
      v2f bf = p[ks * 32];
      acc = __builtin_amdgcn_wmma_f32_16x16x4_f32(false, a, false, bf,
                                                  (short)0, acc, false, false);
    }
#pragma unroll
    for (int v = 0; v < 8; ++v)
      hs[(v + 8 * half) * HS_STRIDE + nt * 16 + col] = fmaxf(acc[v], 0.0f);
  }
  __syncthreads();

  // ============== layer 2: agg2 = adj @ h, hin = agg2/deg =================
#pragma unroll
  for (int nt = 0; nt < 8; ++nt) {
    v8f acc = {};
#pragma unroll
    for (int k = 0; k < 4; ++k) {
      int k0 = 4 * k + 2 * half;
      v2f a  = {adjA[k][0], adjA[k][1]};
      v2f bf = {hs[(k0 + 0) * HS_STRIDE + nt * 16 + col],
                hs[(k0 + 1) * HS_STRIDE + nt * 16 + col]};
      acc = __builtin_amdgcn_wmma_f32_16x16x4_f32(false, a, false, bf,
                                                  (short)0, acc, false, false);
    }
#pragma unroll
    for (int v = 0; v < 8; ++v)
      hin[(v + 8 * half) * HS_STRIDE + nt * 16 + col] = acc[v] * rdeg[v];
  }
  __syncthreads();

  // ========== layer 2 GEMM: out = [hin|h] @ [Wr2;Wo2] + b2 ================
  float* og = out + (size_t)b * NND * FOUT;
  for (int nt = 0; nt < 8; ++nt) {
    float bias = b2[nt * 16 + col];
    v8f acc;
#pragma unroll
    for (int v = 0; v < 8; ++v) acc[v] = bias;
    const v2f* p = PW2 + (nt * 64) * 32 + lane;      // packed B-fragments
    for (int ks = 0; ks < 32; ++ks) {                // K = 128 (neighbor path)
      v2f a  = *(const v2f*)(hin + row * HS_STRIDE + 4 * ks + 2 * half);
      v2f bf = p[ks * 32];
      acc = __builtin_amdgcn_wmma_f32_16x16x4_f32(false, a, false, bf,
                                                  (short)0, acc, false, false);
    }
    for (int ks = 32; ks < 64; ++ks) {               // K = 128 (root path)
      v2f a  = *(const v2f*)(hs + row * HS_STRIDE + 4 * (ks - 32) + 2 * half);
      v2f bf = p[ks * 32];
      acc = __builtin_amdgcn_wmma_f32_16x16x4_f32(false, a, false, bf,
                                                  (short)0, acc, false, false);
    }
#pragma unroll
    for (int v = 0; v < 8; ++v)
      og[(v + 8 * half) * FOUT + nt * 16 + col] = acc[v];
  }
}

extern "C" void kernel_launch(void* const* d_in, const int* in_sizes, int n_in,
                              void* d_out, int out_size, void* d_ws, size_t ws_size,
                              hipStream_t stream) {
  const float* x    = (const float*)d_in[0];
  const int*   ei   = (const int*)d_in[1];
  const float* adjp = (const float*)d_in[2];
  const float* Wr1  = (const float*)d_in[3];
  const float* b1   = (const float*)d_in[4];
  const float* Wo1  = (const float*)d_in[5];
  const float* Wr2  = (const float*)d_in[6];
  const float* b2   = (const float*)d_in[7];
  const float* Wo2  = (const float*)d_in[8];
  float* out = (float*)d_out;

  const int nodes   = in_sizes[0] / FIN;   // B*N = 524288
  const int nedges  = in_sizes[1];         // B*8
  const int batches = nodes / NND;         // 32768

  // workspace layout: flags | PW1 | PW2
  unsigned* flags = (unsigned*)d_ws;
  size_t off = ((size_t)nodes * sizeof(unsigned) + 255) & ~(size_t)255;
  v2f* PW1 = (v2f*)((char*)d_ws + off);
  v2f* PW2 = PW1 + PW1_PAIRS;

  flags_zero_kernel<<<(nodes + 255) / 256, 256, 0, stream>>>(flags, nodes);
  flags_scatter_kernel<<<(nedges + 255) / 256, 256, 0, stream>>>(ei, flags, nedges, nodes);

  const int pack_threads = PW1_PAIRS + PW2_PAIRS;
  pack_weights_kernel<<<(pack_threads + 255) / 256, 256, 0, stream>>>(
      Wr1, Wo1, Wr2, Wo2, PW1, PW2);

  dim3 grid((batches + WAVES - 1) / WAVES);
  sage_fused_kernel<<<grid, WAVES * 32, 0, stream>>>(
      x, adjp, b1, b2, PW1, PW2, flags, out, batches);
}